// AttentionFactorLayer_81037442941477
// MI455X (gfx1250) — compile-verified
//
#include <hip/hip_runtime.h>

// ---------------------------------------------------------------------------
// AttentionFactorLayer for MI455X (gfx1250), fp32 path via V_WMMA_F32_16X16X4_F32
// + TDM (tensor_load_to_lds) staging + non-temporal stores for the 256MB stream.
// N=8192 assets, F=128 features, E=64 embed, K=64 factors.
// ---------------------------------------------------------------------------

typedef float v2f __attribute__((ext_vector_type(2)));
typedef float v4f __attribute__((ext_vector_type(4)));
typedef float v8f __attribute__((ext_vector_type(8)));
typedef unsigned int u32x4 __attribute__((ext_vector_type(4)));
typedef int i32x4 __attribute__((ext_vector_type(4)));
typedef int i32x8 __attribute__((ext_vector_type(8)));

#define NN    8192
#define FF    128
#define EE    64
#define KK    64
#define RIDGE 0.001f
#define NEGINF (-1e9f)

__device__ __forceinline__ v8f wmma_f32(v2f a, v2f b, v8f c) {
    // D = A(16x4 f32) * B(4x16 f32) + C(16x16 f32)
    return __builtin_amdgcn_wmma_f32_16x16x4_f32(false, a, false, b, (short)0, c,
                                                 false, false);
}

// ---------------------------------------------------------------------------
// TDM: DMA a 2D fp32 tile [tile_rows x 128] (row stride = row_stride elements)
// from global memory into LDS at lds_byte_addr. D# per CDNA5 ISA ch.8:
//   group0: count=1 | lds_addr | global_addr(57b) | type=2
//   group1: data_size=4B, tensor_dim0=row_stride, tensor_dim1=tile_rows,
//           tile_dim0=128, tile_dim1=tile_rows, tensor_dim0_stride=row_stride
// Issue once per wave (EXEC ignored by TDM); completion via TENSORcnt.
// ---------------------------------------------------------------------------
__device__ __forceinline__ void tdm_load_tile128(const float* gsrc,
                                                 unsigned lds_byte_addr,
                                                 int tile_rows, int row_stride) {
    unsigned long long ga = (unsigned long long)gsrc;
    u32x4 g0 = {1u,                                       // count=1 (valid D#)
                lds_byte_addr,                            // bits 63:32 lds_addr
                (unsigned)ga,                             // global_addr lo
                (unsigned)((ga >> 32) & 0x01FFFFFFu) | 0x80000000u};  // addr hi | type=2
    i32x8 g1 = {(int)0x00020000,                                      // data_size=2 (4B)
                (int)((row_stride & 0xFFFF) << 16),                   // tensor_dim0 lo16
                (int)(((row_stride >> 16) & 0xFFFF) | ((tile_rows & 0xFFFF) << 16)),
                (int)(((tile_rows >> 16) & 0xFFFF) | (128 << 16)),    // tile_dim0=128
                (int)(tile_rows & 0xFFFF),                            // tile_dim1
                row_stride,                                           // dim0_stride lo32
                0, 0};
    i32x4 z4 = {0, 0, 0, 0};
#if __clang_major__ >= 23
    i32x8 z8 = {0, 0, 0, 0, 0, 0, 0, 0};
    __builtin_amdgcn_tensor_load_to_lds(g0, g1, z4, z4, z8, 0);
#else
    __builtin_amdgcn_tensor_load_to_lds(g0, g1, z4, z4, 0);
#endif
}

// ---------------------------------------------------------------------------
// K0: M = Q @ W_K  -> [64,128]  (tiny, plain VALU)
// ---------------------------------------------------------------------------
__global__ void qwk_kernel(const float* __restrict__ Q, const float* __restrict__ WK,
                           float* __restrict__ Mk) {
    int k = blockIdx.x;   // 0..63
    int f = threadIdx.x;  // 0..127
    float acc = 0.f;
#pragma unroll 8
    for (int e = 0; e < EE; ++e) acc += Q[k * EE + e] * WK[e * FF + f];
    Mk[k * FF + f] = acc;
}

// ---------------------------------------------------------------------------
// K1: scores[k,n] = (sum_f features[n,f]*M[k,f]) / 8, masked. WMMA f32 16x16x4.
// features is a read-once 4MB stream -> non-temporal loads.
// ---------------------------------------------------------------------------
__global__ __launch_bounds__(256) void scores_kernel(const float* __restrict__ feat,
                                                     const int* __restrict__ mask,
                                                     const float* __restrict__ Mk,
                                                     float* __restrict__ out) {
    __shared__ float sF[128][FF];  // features[n0+i][f] : 64KB
    const int n0 = blockIdx.x * 128;
    for (int idx = threadIdx.x; idx < 128 * (FF / 4); idx += 256) {
        int r = idx >> 5, c4 = idx & 31;
        const v4f* src = (const v4f*)(feat + (size_t)(n0 + r) * FF);
        ((v4f*)sF[r])[c4] = __builtin_nontemporal_load(src + c4);
    }
    __syncthreads();

    const int w = threadIdx.x >> 5, lane = threadIdx.x & 31;
    const int hi = lane >> 4, l15 = lane & 15;
    const int kstrip = w & 3, nhalf = w >> 2;
    const int m = kstrip * 16 + l15;  // A-row (factor index)

    for (int nt = 0; nt < 4; ++nt) {
        const int nl = nhalf * 64 + nt * 16 + l15;  // local column
        v8f acc = {0.f, 0.f, 0.f, 0.f, 0.f, 0.f, 0.f, 0.f};
#pragma unroll
        for (int f0 = 0; f0 < FF; f0 += 4) {
            v2f a, b;
            a.x = Mk[m * FF + f0 + 2 * hi];
            a.y = Mk[m * FF + f0 + 2 * hi + 1];
            b.x = sF[nl][f0 + 2 * hi];
            b.y = sF[nl][f0 + 2 * hi + 1];
            acc = wmma_f32(a, b, acc);
        }
        const int n = n0 + nl;
        const int mk = mask[n];
#pragma unroll
        for (int rr = 0; rr < 8; ++rr) {
            int row = kstrip * 16 + rr + 8 * hi;
            out[row * NN + n] = (mk <= 0) ? NEGINF : acc[rr] * 0.125f;
        }
    }
}

// ---------------------------------------------------------------------------
// K2: in-place stable softmax over N per factor + factor_returns.
// ---------------------------------------------------------------------------
__device__ __forceinline__ float waveMax(float v) {
#pragma unroll
    for (int o = 16; o; o >>= 1) v = fmaxf(v, __shfl_xor(v, o, 32));
    return v;
}
__device__ __forceinline__ float waveSum(float v) {
#pragma unroll
    for (int o = 16; o; o >>= 1) v += __shfl_xor(v, o, 32);
    return v;
}
__device__ float blockMax(float v, float* red, int tid) {
    int lane = tid & 31, wid = tid >> 5;
    v = waveMax(v);
    if (lane == 0) red[wid] = v;
    __syncthreads();
    if (wid == 0) {
        float t = red[lane];
        t = waveMax(t);
        if (lane == 0) red[0] = t;
    }
    __syncthreads();
    v = red[0];
    __syncthreads();
    return v;
}
__device__ float blockSum(float v, float* red, int tid) {
    int lane = tid & 31, wid = tid >> 5;
    v = waveSum(v);
    if (lane == 0) red[wid] = v;
    __syncthreads();
    if (wid == 0) {
        float t = red[lane];
        t = waveSum(t);
        if (lane == 0) red[0] = t;
    }
    __syncthreads();
    v = red[0];
    __syncthreads();
    return v;
}

__global__ __launch_bounds__(1024) void softmax_fr_kernel(float* __restrict__ wts,
                                                          const float* __restrict__ rets,
                                                          float* __restrict__ fr) {
    __shared__ float red[32];
    const int k = blockIdx.x, tid = threadIdx.x;
    float s[8], r[8];
    float lm = NEGINF;
#pragma unroll
    for (int i = 0; i < 8; ++i) {
        int n = tid + i * 1024;
        s[i] = wts[k * NN + n];
        r[i] = rets[n];
        lm = fmaxf(lm, s[i]);
    }
    const float vmax = blockMax(lm, red, tid);
    float lsum = 0.f, lwr = 0.f;
#pragma unroll
    for (int i = 0; i < 8; ++i) {
        float e = __expf(s[i] - vmax);
        s[i] = e;
        lsum += e;
        lwr += e * r[i];
    }
    const float tsum = blockSum(lsum, red, tid);
    const float twr = blockSum(lwr, red, tid);
    const float inv = 1.0f / tsum;
#pragma unroll
    for (int i = 0; i < 8; ++i) wts[k * NN + tid + i * 1024] = s[i] * inv;
    if (tid == 0) fr[k] = twr * inv;
}

// ---------------------------------------------------------------------------
// K3a/K3: gram = W @ W^T + RIDGE*I.  Split-N WMMA partials + f32 atomics.
// ---------------------------------------------------------------------------
__global__ void graminit_kernel(float* __restrict__ gram) {
    int i = blockIdx.x * 256 + threadIdx.x;
    if (i < KK * KK) gram[i] = ((i >> 6) == (i & 63)) ? RIDGE : 0.f;
}

__global__ __launch_bounds__(256) void gram_kernel(const float* __restrict__ wts,
                                                   float* __restrict__ gram) {
    __shared__ float sW[KK][256];  // 64KB chunk of W columns
    const int n0 = blockIdx.x * 256;
    for (int idx = threadIdx.x; idx < KK * 64; idx += 256) {
        int r = idx >> 6, c4 = idx & 63;
        ((v4f*)sW[r])[c4] = ((const v4f*)(wts + (size_t)r * NN + n0))[c4];
    }
    __syncthreads();

    const int w = threadIdx.x >> 5, lane = threadIdx.x & 31;
    const int hi = lane >> 4, l15 = lane & 15;
    for (int t = w * 2; t < w * 2 + 2; ++t) {  // 16 output tiles, 2 per wave
        const int ti = t >> 2, tj = t & 3;
        const int ra = ti * 16 + l15, rb = tj * 16 + l15;
        v8f acc = {0.f, 0.f, 0.f, 0.f, 0.f, 0.f, 0.f, 0.f};
#pragma unroll
        for (int nb = 0; nb < 256; nb += 4) {
            v2f a, b;
            a.x = sW[ra][nb + 2 * hi];
            a.y = sW[ra][nb + 2 * hi + 1];
            b.x = sW[rb][nb + 2 * hi];
            b.y = sW[rb][nb + 2 * hi + 1];
            acc = wmma_f32(a, b, acc);
        }
#pragma unroll
        for (int rr = 0; rr < 8; ++rr)
            atomicAdd(&gram[(ti * 16 + rr + 8 * hi) * KK + tj * 16 + l15], acc[rr]);
    }
}

// ---------------------------------------------------------------------------
// K5: 64x64 Gauss-Jordan inversion (SPD + ridge -> no pivoting), one block.
// ---------------------------------------------------------------------------
__global__ __launch_bounds__(1024) void inv_kernel(const float* __restrict__ gram,
                                                   float* __restrict__ invout) {
    __shared__ float aug[KK][128];  // [A | I], 32KB
    const int tid = threadIdx.x;
    for (int idx = tid; idx < KK * 128; idx += 1024) {
        int r = idx >> 7, c = idx & 127;
        aug[r][c] = (c < 64) ? gram[r * KK + c] : ((c - 64) == r ? 1.f : 0.f);
    }
    __syncthreads();
    const int r = tid >> 4;
    const int cbase = (tid & 15) * 8;
    for (int p = 0; p < KK; ++p) {
        const float pinv = 1.0f / aug[p][p];
        __syncthreads();
        if (tid < 128) aug[p][tid] *= pinv;
        __syncthreads();
        const float factor = aug[r][p];
        __syncthreads();
        if (r != p) {
#pragma unroll
            for (int c = cbase; c < cbase + 8; ++c) aug[r][c] -= factor * aug[p][c];
        }
        __syncthreads();
    }
    for (int idx = tid; idx < KK * KK; idx += 1024)
        invout[idx] = aug[idx >> 6][64 + (idx & 63)];
}

// ---------------------------------------------------------------------------
// K6: betaT (= beta_t.T, [64,N]) = inv @ W   (inv symmetric). WMMA.
// ---------------------------------------------------------------------------
__global__ __launch_bounds__(256) void beta_kernel(const float* __restrict__ invm,
                                                   const float* __restrict__ wts,
                                                   float* __restrict__ betaT) {
    __shared__ float sW[KK][128];  // 32KB
    __shared__ float sI[KK][KK];   // 16KB
    const int n0 = blockIdx.x * 128;
    for (int idx = threadIdx.x; idx < KK * 32; idx += 256) {
        int r = idx >> 5, c4 = idx & 31;
        ((v4f*)sW[r])[c4] = ((const v4f*)(wts + (size_t)r * NN + n0))[c4];
    }
    for (int idx = threadIdx.x; idx < KK * 16; idx += 256) {
        int r = idx >> 4, c4 = idx & 15;
        ((v4f*)sI[r])[c4] = ((const v4f*)(invm + r * KK))[c4];
    }
    __syncthreads();

    const int w = threadIdx.x >> 5, lane = threadIdx.x & 31;
    const int hi = lane >> 4, l15 = lane & 15;
    const int strip = w & 3, nhalf = w >> 2;
    for (int nt = 0; nt < 4; ++nt) {
        const int nl = nhalf * 64 + nt * 16 + l15;
        v8f acc = {0.f, 0.f, 0.f, 0.f, 0.f, 0.f, 0.f, 0.f};
#pragma unroll
        for (int kk = 0; kk < KK; kk += 4) {
            v2f a, b;
            a.x = sI[strip * 16 + l15][kk + 2 * hi];
            a.y = sI[strip * 16 + l15][kk + 2 * hi + 1];
            b.x = sW[kk + 2 * hi][nl];
            b.y = sW[kk + 2 * hi + 1][nl];
            acc = wmma_f32(a, b, acc);
        }
#pragma unroll
        for (int rr = 0; rr < 8; ++rr)
            betaT[(strip * 16 + rr + 8 * hi) * NN + n0 + nl] = acc[rr];
    }
}

// ---------------------------------------------------------------------------
// K4: predicted = W^T @ fr ; residual = returns - betaT^T @ fr
// ---------------------------------------------------------------------------
__global__ __launch_bounds__(256) void predres_kernel(const float* __restrict__ wts,
                                                      const float* __restrict__ betaT,
                                                      const float* __restrict__ fr,
                                                      const float* __restrict__ rets,
                                                      float* __restrict__ predicted,
                                                      float* __restrict__ residual) {
    __shared__ float sfr[KK];
    if (threadIdx.x < KK) sfr[threadIdx.x] = fr[threadIdx.x];
    __syncthreads();
    const int n = blockIdx.x * 256 + threadIdx.x;
    float p = 0.f, rr = 0.f;
#pragma unroll 8
    for (int k = 0; k < KK; ++k) {
        const float f = sfr[k];
        p += wts[k * NN + n] * f;
        rr += betaT[k * NN + n] * f;
    }
    predicted[n] = p;
    residual[n] = rets[n] - rr;
}

// ---------------------------------------------------------------------------
// K7: omega[n,m] = (n==m) - sum_k betaT[k,n]*W[k,m].  8192x8192 WMMA GEMM.
// TDM stages both 64x128 operand tiles into LDS (TENSORcnt-tracked); output is
// a write-once 256MB stream -> non-temporal stores (keep L2 for W/betaT reuse).
// ---------------------------------------------------------------------------
__global__ __launch_bounds__(256) void omega_kernel(const float* __restrict__ betaT,
                                                    const float* __restrict__ wts,
                                                    float* __restrict__ omega) {
    __shared__ float sB[KK][128];  // betaT[:, n-tile]  32KB
    __shared__ float sW[KK][128];  // W[:, m-tile]      32KB
    const int mb0 = blockIdx.x * 128;
    const int nb0 = blockIdx.y * 128;

    if (threadIdx.x < 32) {  // wave 0 issues both tile DMAs
        tdm_load_tile128(betaT + nb0, (unsigned)(unsigned long long)&sB[0][0], KK, NN);
        tdm_load_tile128(wts + mb0, (unsigned)(unsigned long long)&sW[0][0], KK, NN);
        __builtin_amdgcn_s_wait_tensorcnt(0);
    }
    // Escape the LDS buffers so the compiler treats them as written by the TDM.
    asm volatile("" ::"v"(&sB[0][0]), "v"(&sW[0][0]) : "memory");
    __syncthreads();

    const int w = threadIdx.x >> 5, lane = threadIdx.x & 31;
    const int hi = lane >> 4, l15 = lane & 15;
    const int al = w * 16 + l15;  // local output row (n)
    for (int mt = 0; mt < 8; ++mt) {
        const int ml = mt * 16 + l15;  // local output col (m)
        v8f acc = {0.f, 0.f, 0.f, 0.f, 0.f, 0.f, 0.f, 0.f};
#pragma unroll
        for (int kk = 0; kk < KK; kk += 4) {
            v2f a, b;
            a.x = sB[kk + 2 * hi][al];
            a.y = sB[kk + 2 * hi + 1][al];
            b.x = sW[kk + 2 * hi][ml];
            b.y = sW[kk + 2 * hi + 1][ml];
            acc = wmma_f32(a, b, acc);
        }
#pragma unroll
        for (int rr = 0; rr < 8; ++rr) {
            const int nrow = nb0 + w * 16 + rr + 8 * hi;
            const int mcol = mb0 + ml;
            const float v = ((nrow == mcol) ? 1.0f : 0.0f) - acc[rr];
            __builtin_nontemporal_store(v, &omega[(size_t)nrow * NN + mcol]);
        }
    }
}

// ---------------------------------------------------------------------------
extern "C" void kernel_launch(void* const* d_in, const int* in_sizes, int n_in,
                              void* d_out, int out_size, void* d_ws, size_t ws_size,
                              hipStream_t stream) {
    const float* features = (const float*)d_in[0];  // [8192,128]
    const float* returns  = (const float*)d_in[1];  // [8192,1]
    const int*   mask     = (const int*)d_in[2];    // [8192]
    const float* W_K      = (const float*)d_in[3];  // [64,128]
    const float* Q        = (const float*)d_in[4];  // [64,64]

    float* out       = (float*)d_out;
    float* o_weights = out;                      // 524288
    float* o_fr      = out + 524288;             // 64
    float* o_pred    = out + 524352;             // 8192
    float* o_res     = out + 532544;             // 8192
    float* o_omega   = out + 540736;             // 67108864
    float* o_betaT   = out + 540736 + 67108864;  // 524288

    float* ws      = (float*)d_ws;
    float* ws_M    = ws;          // 8192 floats  (Q @ W_K)
    float* ws_gram = ws + 8192;   // 4096 floats
    float* ws_inv  = ws + 12288;  // 4096 floats

    // 1) M = Q @ W_K
    qwk_kernel<<<64, 128, 0, stream>>>(Q, W_K, ws_M);
    // 2) scores (masked, scaled) -> weights slot
    scores_kernel<<<NN / 128, 256, 0, stream>>>(features, mask, ws_M, o_weights);
    // 3) softmax in-place + factor_returns
    softmax_fr_kernel<<<KK, 1024, 0, stream>>>(o_weights, returns, o_fr);
    // 4) gram = W W^T + ridge I
    graminit_kernel<<<16, 256, 0, stream>>>(ws_gram);
    gram_kernel<<<NN / 256, 256, 0, stream>>>(o_weights, ws_gram);
    // 5) inv(gram)
    inv_kernel<<<1, 1024, 0, stream>>>(ws_gram, ws_inv);
    // 6) betaT = inv @ W
    beta_kernel<<<NN / 128, 256, 0, stream>>>(ws_inv, o_weights, o_betaT);
    // 7) predicted & residual (residual collapsed algebraically)
    predres_kernel<<<NN / 256, 256, 0, stream>>>(o_weights, o_betaT, o_fr, returns,
                                                 o_pred, o_res);
    // 8) omega = I - betaT^T @ W  (the 256MB GEMM)
    omega_kernel<<<dim3(NN / 128, NN / 128), 256, 0, stream>>>(o_betaT, o_weights,
                                                               o_omega);
}